// SGDStep_20349555048651
// MI455X (gfx1250) — compile-verified
//
#include <hip/hip_runtime.h>

// N=400000 independent 5->8->8->2 MLPs, one SGD step each.
// Memory-bound: ~1.13KB traffic / ~600 FLOP per sample (AI ~0.5 F/B).
// Strategy: TDM (tensor_load_to_lds / tensor_store_from_lds) streams each
// block's contiguous 35KB weight tile HBM<->LDS; fp32 VALU math per thread.

#define LRATE 0.01f
#define SPB 64          // samples per block == threads per block (2 waves, wave32)
#define ROW 138         // floats per sample row

static_assert(SPB * ROW < 65536, "tile_dim0 must fit in 16 bits");

typedef unsigned int u32x4 __attribute__((ext_vector_type(4)));
typedef int i32x4 __attribute__((ext_vector_type(4)));
typedef int i32x8 __attribute__((ext_vector_type(8)));
typedef __attribute__((address_space(3))) float lds_f;

__device__ __forceinline__ float fast_sigmoid(float z) {
  // sigmoid(z) = rcp(1 + exp2(-z*log2(e)))  -> v_exp_f32 + v_rcp_f32
  float e = __builtin_amdgcn_exp2f(z * -1.44269504088896340736f);
  return __builtin_amdgcn_rcpf(1.0f + e);
}

__device__ __forceinline__ void wait_tensor0() {
#if __has_builtin(__builtin_amdgcn_s_wait_tensorcnt)
  __builtin_amdgcn_s_wait_tensorcnt(0);
#else
  asm volatile("s_wait_tensorcnt 0x0" ::: "memory");
#endif
}

#if __has_builtin(__builtin_amdgcn_tensor_load_to_lds) && \
    __has_builtin(__builtin_amdgcn_tensor_store_from_lds)
#define USE_TDM 1
#else
#define USE_TDM 0
#endif

#if USE_TDM
// Build a 1-D TDM descriptor: copy `nelem` fp32 elements between a
// contiguous global region and LDS.
__device__ __forceinline__ void tdm_make_desc(unsigned long long gaddr,
                                              unsigned lds_addr, unsigned nelem,
                                              u32x4& g0, i32x8& g1) {
  // ---- group 0 (128b) ----
  g0[0] = 1u;                                     // count=1 (valid), no gather
  g0[1] = lds_addr;                               // lds_addr (bytes)
  g0[2] = (unsigned)(gaddr & 0xffffffffull);      // global_addr[31:0]
  g0[3] = (unsigned)((gaddr >> 32) & 0x01ffffffull) // global_addr[56:32]
          | 0x80000000u;                          // type=2 ("image")
  // ---- group 1 (256b) ----
  g1[0] = 0x00020000;                             // wg_mask=0, data_size=2 (4B)
  g1[1] = (int)((nelem & 0xffffu) << 16);         // tensor_dim0[15:0] @ bits 63:48
  g1[2] = (int)(((nelem >> 16) & 0xffffu)         // tensor_dim0[31:16]
          | (1u << 16));                          // tensor_dim1 = 1
  g1[3] = (int)((nelem & 0xffffu) << 16);         // tile_dim0 = nelem (1-D tile)
  g1[4] = 0;                                      // tile_dim1=0, tile_dim2=0 (unused)
  g1[5] = (int)nelem;                             // tensor_dim0_stride[31:0]
  g1[6] = 0;                                      // stride0[47:32]=0, stride1 lo=0
  g1[7] = 0;                                      // stride1 hi
}

__device__ __forceinline__ void tdm_load(unsigned long long gaddr,
                                         unsigned lds_addr, unsigned nelem) {
  u32x4 g0; i32x8 g1;
  tdm_make_desc(gaddr, lds_addr, nelem, g0, g1);
  i32x4 z4 = {0, 0, 0, 0};
#if defined(__clang_major__) && (__clang_major__ >= 23)
  i32x8 z8 = {0, 0, 0, 0, 0, 0, 0, 0};
  __builtin_amdgcn_tensor_load_to_lds(g0, g1, z4, z4, z8, 0);
#else
  __builtin_amdgcn_tensor_load_to_lds(g0, g1, z4, z4, 0);
#endif
}

__device__ __forceinline__ void tdm_store(unsigned long long gaddr,
                                          unsigned lds_addr, unsigned nelem) {
  u32x4 g0; i32x8 g1;
  tdm_make_desc(gaddr, lds_addr, nelem, g0, g1);
  i32x4 z4 = {0, 0, 0, 0};
#if defined(__clang_major__) && (__clang_major__ >= 23)
  i32x8 z8 = {0, 0, 0, 0, 0, 0, 0, 0};
  __builtin_amdgcn_tensor_store_from_lds(g0, g1, z4, z4, z8, 0);
#else
  __builtin_amdgcn_tensor_store_from_lds(g0, g1, z4, z4, 0);
#endif
}
#endif  // USE_TDM

__global__ __launch_bounds__(SPB) void mlp_sgd_tdm(
    const float* __restrict__ weights, const float* __restrict__ x,
    const float* __restrict__ y, float* __restrict__ out, int ntotal) {
  __shared__ float sw[SPB * ROW];  // 35,328 B: one block's weight tile

  const int t = threadIdx.x;
  const long long s0 = (long long)blockIdx.x * SPB;  // first sample of block
  int valid = ntotal - (int)s0;
  if (valid > SPB) valid = SPB;
  const unsigned nelem = (unsigned)valid * ROW;
  const long long base = s0 * ROW;

  // ---- stage weights tile HBM -> LDS ----
#if USE_TDM
  if (t == 0) {  // single-lane branch: only wave 0 reaches the TDM issue
    unsigned lds_addr = (unsigned)(unsigned long long)(lds_f*)&sw[0];
    tdm_load((unsigned long long)(const void*)(weights + base), lds_addr, nelem);
  }
  wait_tensor0();
#else
  for (int i = t; i < (int)nelem; i += SPB) sw[i] = weights[base + i];
#endif
  __syncthreads();
  asm volatile("" ::: "memory");  // LDS now holds DMA-written data

  // ---- per-sample fwd/bwd/update, all in fp32 VALU out of LDS ----
  if (t < valid) {
    float* w = &sw[t * ROW];
    const long long n = s0 + t;

    float xv[5];
#pragma unroll
    for (int d = 0; d < 5; ++d) xv[d] = x[n * 5 + d];
    const float y0 = y[n * 2 + 0];
    const float y1 = y[n * 2 + 1];

    // layer 1: z1 = x@W1 + b1 ; a1 = relu(z1)     (W1[d][h] = w[d*8+h])
    float z1[8], a1[8];
#pragma unroll
    for (int h = 0; h < 8; ++h) {
      float acc = w[40 + h];
#pragma unroll
      for (int d = 0; d < 5; ++d) acc += xv[d] * w[d * 8 + h];
      z1[h] = acc;
      a1[h] = acc > 0.0f ? acc : 0.0f;
    }

    // layer 2: a2 = sigmoid(a1@W2 + b2)           (W2[h][k] = w[48+h*8+k])
    float a2[8];
#pragma unroll
    for (int k = 0; k < 8; ++k) {
      float acc = w[112 + k];
#pragma unroll
      for (int h = 0; h < 8; ++h) acc += a1[h] * w[48 + h * 8 + k];
      a2[k] = fast_sigmoid(acc);
    }

    // layer 3: z3 = a2@W3 + b3 ; d_out = z3 - y   (W3[k][o] = w[120+k*2+o])
    float dout[2];
#pragma unroll
    for (int o = 0; o < 2; ++o) {
      float acc = w[136 + o];
#pragma unroll
      for (int k = 0; k < 8; ++k) acc += a2[k] * w[120 + k * 2 + o];
      dout[o] = acc - (o == 0 ? y0 : y1);
    }

    // d_z2 = (d_out@W3^T) * a2 * (1-a2)   (uses OLD W3)
    float dz2[8];
#pragma unroll
    for (int k = 0; k < 8; ++k) {
      float da2 = dout[0] * w[120 + k * 2] + dout[1] * w[120 + k * 2 + 1];
      dz2[k] = da2 * a2[k] * (1.0f - a2[k]);
    }
    // update W3, b3 (old W3 consumed above)
#pragma unroll
    for (int k = 0; k < 8; ++k) {
      w[120 + k * 2 + 0] -= LRATE * (a2[k] * dout[0]);
      w[120 + k * 2 + 1] -= LRATE * (a2[k] * dout[1]);
    }
    w[136] -= LRATE * dout[0];
    w[137] -= LRATE * dout[1];

    // d_z1 = (d_z2@W2^T) * (z1>0)        (uses OLD W2)
    float dz1[8];
#pragma unroll
    for (int h = 0; h < 8; ++h) {
      float da1 = 0.0f;
#pragma unroll
      for (int k = 0; k < 8; ++k) da1 += dz2[k] * w[48 + h * 8 + k];
      dz1[h] = z1[h] > 0.0f ? da1 : 0.0f;
    }
    // update W2, b2
#pragma unroll
    for (int h = 0; h < 8; ++h)
#pragma unroll
      for (int k = 0; k < 8; ++k)
        w[48 + h * 8 + k] -= LRATE * (a1[h] * dz2[k]);
#pragma unroll
    for (int k = 0; k < 8; ++k) w[112 + k] -= LRATE * dz2[k];

    // update W1, b1
#pragma unroll
    for (int d = 0; d < 5; ++d)
#pragma unroll
      for (int h = 0; h < 8; ++h)
        w[d * 8 + h] -= LRATE * (xv[d] * dz1[h]);
#pragma unroll
    for (int h = 0; h < 8; ++h) w[40 + h] -= LRATE * dz1[h];
  }

  __syncthreads();
  asm volatile("" ::: "memory");  // updated tile committed to LDS

  // ---- stream updated tile LDS -> HBM (d_out) ----
#if USE_TDM
  if (t == 0) {
    unsigned lds_addr = (unsigned)(unsigned long long)(lds_f*)&sw[0];
    tdm_store((unsigned long long)(void*)(out + base), lds_addr, nelem);
  }
  wait_tensor0();
#else
  for (int i = t; i < (int)nelem; i += SPB) out[base + i] = sw[i];
#endif
}

extern "C" void kernel_launch(void* const* d_in, const int* in_sizes, int n_in,
                              void* d_out, int out_size, void* d_ws, size_t ws_size,
                              hipStream_t stream) {
  (void)n_in; (void)out_size; (void)d_ws; (void)ws_size;
  const float* weights = (const float*)d_in[0];  // (N,138) f32
  const float* x = (const float*)d_in[1];        // (N,5)   f32
  const float* y = (const float*)d_in[2];        // (N,2)   f32
  float* out = (float*)d_out;                    // (N,138) f32

  const int ntotal = in_sizes[0] / ROW;          // N = 400000
  const int blocks = (ntotal + SPB - 1) / SPB;   // 6250
  mlp_sgd_tdm<<<blocks, SPB, 0, stream>>>(weights, x, y, out, ntotal);
}